// StudentT_KSD_10307921510998
// MI455X (gfx1250) — compile-verified
//
#include <hip/hip_runtime.h>
#include <stdint.h>

typedef __attribute__((ext_vector_type(2))) float v2f;
typedef __attribute__((ext_vector_type(8))) float v8f;

#define D_DIM 64
#define TPB 256

// ---------------------------------------------------------------------------
// Kernel 1: G = Z * Z^T via V_WMMA_F32_16X16X4_F32.
// One wave (32 lanes) computes one 16x16 tile of G; K=64 -> 16 WMMA steps.
// A-frag (16x4 f32): lane m (<16): v0=A[m][k0], v1=A[m][k0+1];
//                    lane m+16   : v0=A[m][k0+2], v1=A[m][k0+3].
// B-frag (4x16) mirrors this with N = lane index (B[k][n] = Z[col0+n][k]).
// C/D (16x16 f32): VGPR r, lanes 0-15 -> M=r, N=lane; lanes 16-31 -> M=r+8.
// ---------------------------------------------------------------------------
__global__ __launch_bounds__(128) void gram_wmma(const float* __restrict__ Z,
                                                 float* __restrict__ G, int n) {
  const int lane = threadIdx.x & 31;
  const int wave = threadIdx.x >> 5;
  const int ntiles = n >> 4;
  const int tile = blockIdx.x * 4 + wave;          // wave-uniform
  if (tile >= ntiles * ntiles) return;             // uniform exit, EXEC stays full
  const int ti = tile / ntiles;
  const int tj = tile - ti * ntiles;

  const int half = lane >> 4;        // 0: K-pair {0,1}; 1: K-pair {2,3}
  const int m    = lane & 15;        // M (for A) / N (for B)

  const float* arow = Z + (size_t)(ti * 16 + m) * D_DIM;
  const float* brow = Z + (size_t)(tj * 16 + m) * D_DIM;

  v2f a[16], b[16];
#pragma unroll
  for (int kk = 0; kk < 16; ++kk) {
    const int k = kk * 4 + 2 * half;
    a[kk] = *(const v2f*)(arow + k);
    b[kk] = *(const v2f*)(brow + k);
  }

  v8f c = {};
#pragma unroll
  for (int kk = 0; kk < 16; ++kk) {
    // 8 args: (neg_a, A, neg_b, B, c_mod, C, reuse_a, reuse_b)
    c = __builtin_amdgcn_wmma_f32_16x16x4_f32(false, a[kk], false, b[kk],
                                              (short)0, c, false, false);
  }

#pragma unroll
  for (int r = 0; r < 8; ++r) {
    const int row = ti * 16 + r + 8 * half;
    const int col = tj * 16 + m;
    G[(size_t)row * n + col] = c[r];
  }
}

// ---------------------------------------------------------------------------
// Kernel 2: row norms ||z_i||^2 and Student-t score scale c_i.
// ---------------------------------------------------------------------------
__global__ __launch_bounds__(TPB) void row_stats(const float* __restrict__ Z,
                                                 float* __restrict__ norms,
                                                 float* __restrict__ cvec, int n) {
  const int i = blockIdx.x * blockDim.x + threadIdx.x;
  if (i >= n) return;
  const float* zr = Z + (size_t)i * D_DIM;
  float s = 0.0f;
#pragma unroll
  for (int k = 0; k < D_DIM; ++k) s = fmaf(zr[k], zr[k], s);
  norms[i] = s;
  // c_i = -(nu + d) / (nu*sigma^2 + ||z_i||^2),  nu=3, sigma=1, d=64
  cvec[i] = -(3.0f + (float)D_DIM) / (3.0f + s);
}

// ---------------------------------------------------------------------------
// Exact median of dist_sq over all n^2 entries (matches jnp.median on even
// count: mean of order statistics k1=N/2 and k2=N/2+1, 1-indexed).
// Bisection on uint bit patterns (valid since dist_sq >= 0).
// State layout (uint32): [0]=lo1 [1]=hi1 [2]=mid1 [3]=cnt1
//                        [4]=lo2 [5]=hi2 [6]=mid2 [7]=cnt2  [8]=alpha(bits)
// ---------------------------------------------------------------------------
__global__ void med_init(uint32_t* st) {
  if (threadIdx.x == 0 && blockIdx.x == 0) {
    st[0] = 0u; st[1] = 0x7F800000u; st[2] = st[0] + ((st[1] - st[0]) >> 1); st[3] = 0u;
    st[4] = 0u; st[5] = 0x7F800000u; st[6] = st[4] + ((st[5] - st[4]) >> 1); st[7] = 0u;
  }
}

__global__ __launch_bounds__(TPB) void med_count(const float* __restrict__ G,
                                                 const float* __restrict__ norms,
                                                 uint32_t* __restrict__ st, int n) {
  __shared__ unsigned sh1[TPB];
  __shared__ unsigned sh2[TPB];
  const size_t nn  = (size_t)n * n;
  const size_t idx = (size_t)blockIdx.x * blockDim.x + threadIdx.x;
  const unsigned mid1 = st[2];
  const unsigned mid2 = st[6];
  unsigned p1 = 0, p2 = 0;
  if (idx < nn) {
    const int i = (int)(idx / n);
    const int j = (int)(idx - (size_t)i * n);
    const float dsq = fmaxf(norms[i] + norms[j] - 2.0f * G[idx], 0.0f);
    const unsigned bits = __float_as_uint(dsq);
    p1 = (bits <= mid1) ? 1u : 0u;
    p2 = (bits <= mid2) ? 1u : 0u;
  }
  sh1[threadIdx.x] = p1;
  sh2[threadIdx.x] = p2;
  __syncthreads();
  for (int s = TPB / 2; s > 0; s >>= 1) {
    if (threadIdx.x < s) {
      sh1[threadIdx.x] += sh1[threadIdx.x + s];
      sh2[threadIdx.x] += sh2[threadIdx.x + s];
    }
    __syncthreads();
  }
  if (threadIdx.x == 0) {
    atomicAdd(&st[3], sh1[0]);
    atomicAdd(&st[7], sh2[0]);
  }
}

__global__ void med_update(uint32_t* st, int final_it, unsigned k1, unsigned k2) {
  if (threadIdx.x == 0 && blockIdx.x == 0) {
    if (st[3] >= k1) st[1] = st[2]; else st[0] = st[2] + 1u;
    st[2] = st[0] + ((st[1] - st[0]) >> 1);
    st[3] = 0u;
    if (st[7] >= k2) st[5] = st[6]; else st[4] = st[6] + 1u;
    st[6] = st[4] + ((st[5] - st[4]) >> 1);
    st[7] = 0u;
    if (final_it) {
      const float v1 = __uint_as_float(st[0]);
      const float v2 = __uint_as_float(st[4]);
      const float med = 0.5f * (v1 + v2);
      ((float*)st)[8] = 1.0f / (med + 1e-6f);  // alpha
    }
  }
}

// ---------------------------------------------------------------------------
// Kernel: h_ij from closed form; deterministic block-tree partial sums.
//   g=G_ij, base=1+a*dsq, K=base^-1/2, gc=-a*K^3
//   term_a = c_i c_j g K
//   term_b = -gc * c_i * (G_ii - g)
//   term_c =  gc * c_j * (g - G_jj)
//   lap    =  gc * (d - 3 a dsq / base)
// ---------------------------------------------------------------------------
__global__ __launch_bounds__(TPB) void ksd_sum(const float* __restrict__ G,
                                               const float* __restrict__ norms,
                                               const float* __restrict__ cvec,
                                               const uint32_t* __restrict__ st,
                                               float* __restrict__ partials, int n) {
  __shared__ float sh[TPB];
  const float alpha = ((const float*)st)[8];
  const size_t nn  = (size_t)n * n;
  const size_t idx = (size_t)blockIdx.x * blockDim.x + threadIdx.x;
  float h = 0.0f;
  if (idx < nn) {
    const int i = (int)(idx / n);
    const int j = (int)(idx - (size_t)i * n);
    if (i != j) {
      const float g  = G[idx];
      const float ni = norms[i], nj = norms[j];
      const float ci = cvec[i],  cj = cvec[j];
      const float dsq  = fmaxf(ni + nj - 2.0f * g, 0.0f);
      const float base = fmaf(alpha, dsq, 1.0f);
      const float Kk   = 1.0f / sqrtf(base);        // base^{-1/2}
      const float gc   = -alpha * (Kk * Kk * Kk);   // -2*a*beta*base^{-3/2}, beta=1/2
      const float term_a = ci * cj * g * Kk;
      const float term_b = -gc * ci * (ni - g);
      const float term_c =  gc * cj * (g - nj);
      const float lap    =  gc * ((float)D_DIM - 3.0f * alpha * dsq / base);
      h = term_a + term_b + term_c + lap;
    }
  }
  sh[threadIdx.x] = h;
  __syncthreads();
  for (int s = TPB / 2; s > 0; s >>= 1) {
    if (threadIdx.x < s) sh[threadIdx.x] += sh[threadIdx.x + s];
    __syncthreads();
  }
  if (threadIdx.x == 0) partials[blockIdx.x] = sh[0];
}

__global__ __launch_bounds__(TPB) void final_reduce(const float* __restrict__ partials,
                                                    int nparts, float* __restrict__ out,
                                                    float scale) {
  __shared__ float sh[TPB];
  float acc = 0.0f;
  for (int p = threadIdx.x; p < nparts; p += TPB) acc += partials[p];  // fixed order
  sh[threadIdx.x] = acc;
  __syncthreads();
  for (int s = TPB / 2; s > 0; s >>= 1) {
    if (threadIdx.x < s) sh[threadIdx.x] += sh[threadIdx.x + s];
    __syncthreads();
  }
  if (threadIdx.x == 0) out[0] = sh[0] * scale;
}

// ---------------------------------------------------------------------------
extern "C" void kernel_launch(void* const* d_in, const int* in_sizes, int n_in,
                              void* d_out, int out_size, void* d_ws, size_t ws_size,
                              hipStream_t stream) {
  const float* Z = (const float*)d_in[0];
  const int n = in_sizes[0] / D_DIM;          // 1536
  const size_t nn = (size_t)n * n;

  // Workspace layout: G (nn f32) | norms (n) | c (n) | state (16 u32) | partials
  float*    G        = (float*)d_ws;
  float*    norms    = G + nn;
  float*    cvec     = norms + n;
  uint32_t* st       = (uint32_t*)(cvec + n);
  float*    partials = (float*)(st + 16);

  const int ntiles    = n / 16;
  const int tileCount = ntiles * ntiles;
  const int gemmBlks  = (tileCount + 3) / 4;           // 4 waves/block
  const int nblocks   = (int)((nn + TPB - 1) / TPB);   // elementwise grid

  gram_wmma<<<gemmBlks, 128, 0, stream>>>(Z, G, n);
  row_stats<<<(n + TPB - 1) / TPB, TPB, 0, stream>>>(Z, norms, cvec, n);

  med_init<<<1, 32, 0, stream>>>(st);
  const unsigned k1 = (unsigned)(nn / 2);        // 1-indexed order statistics
  const unsigned k2 = (unsigned)(nn / 2 + 1);
  for (int it = 0; it < 32; ++it) {
    med_count<<<nblocks, TPB, 0, stream>>>(G, norms, st, n);
    med_update<<<1, 32, 0, stream>>>(st, it == 31 ? 1 : 0, k1, k2);
  }

  ksd_sum<<<nblocks, TPB, 0, stream>>>(G, norms, cvec, st, partials, n);
  const float scale = 1.0f / ((float)n * (float)(n - 1));
  final_reduce<<<1, TPB, 0, stream>>>(partials, nblocks, (float*)d_out, scale);
}